// DotPredictor_30691836297942
// MI455X (gfx1250) — compile-verified
//
#include <hip/hip_runtime.h>

// u_dot_v edge scoring: out[e] = dot(h[src[e]], h[dst[e]]), D=128, f32.
// Strategy: 16 edges per wave, S = HU * HV^T via chained V_WMMA_F32_16X16X4_F32,
// take diag(S). Memory-bound gather; h (51 MB) is L2-resident on MI455X (192 MB L2).

typedef __attribute__((ext_vector_type(2))) float v2f;
typedef __attribute__((ext_vector_type(8))) float v8f;

#define D_FEAT 128
#define WAVES_PER_BLOCK 8
#define EDGES_PER_WAVE 16

__global__ __launch_bounds__(256) void edge_dot_wmma_f32(
    const float* __restrict__ h,
    const int*   __restrict__ src,
    const int*   __restrict__ dst,
    float*       __restrict__ out,
    int n_edges)
{
    const int lane = threadIdx.x & 31;
    const int wave = threadIdx.x >> 5;
    const int tile = blockIdx.x * WAVES_PER_BLOCK + wave;  // 16 edges / tile
    const int e0   = tile * EDGES_PER_WAVE;
    if (e0 >= n_edges) return;            // wave-uniform: EXEC stays all-ones

    const int row  = lane & 15;           // edge within tile
    const int half = lane >> 4;           // K-half for f32 WMMA A/B layout

    // Clamp so a (hypothetical) partial tile still loads valid rows.
    int e  = e0 + row;
    int ec = e < n_edges ? e : n_edges - 1;

    const int su = src[ec];
    const int sv = dst[ec];

    // Per-lane base: lane m   -> feats k+0,k+1 of its row
    //               lane m+16 -> feats k+2,k+3 of its row
    // This is exactly the V_WMMA_F32_16X16X4_F32 A (16x4) and B (4x16) layout.
    const float* pu = h + (long)su * D_FEAT + 2 * half;
    const float* pv = h + (long)sv * D_FEAT + 2 * half;

    // Prefetch next tile's index data (global_prefetch_b8; uniform branch).
    if (e0 + EDGES_PER_WAVE < n_edges) {
        __builtin_prefetch(&src[e0 + EDGES_PER_WAVE + row], 0, 1);
        __builtin_prefetch(&dst[e0 + EDGES_PER_WAVE + row], 0, 1);
    }

    v8f acc0 = {0.f, 0.f, 0.f, 0.f, 0.f, 0.f, 0.f, 0.f};
    v8f acc1 = {0.f, 0.f, 0.f, 0.f, 0.f, 0.f, 0.f, 0.f};

    // 32 chained WMMAs over K=128, two accumulators for ILP.
#pragma unroll
    for (int k = 0; k < D_FEAT; k += 8) {
        v2f a0 = *(const v2f*)(pu + k);
        v2f b0 = *(const v2f*)(pv + k);
        v2f a1 = *(const v2f*)(pu + k + 4);
        v2f b1 = *(const v2f*)(pv + k + 4);
        // 8 args: (neg_a, A, neg_b, B, c_mod, C, reuse_a, reuse_b)
        acc0 = __builtin_amdgcn_wmma_f32_16x16x4_f32(
            false, a0, false, b0, (short)0, acc0, false, false);
        acc1 = __builtin_amdgcn_wmma_f32_16x16x4_f32(
            false, a1, false, b1, (short)0, acc1, false, false);
    }
    v8f c = acc0 + acc1;   // summing two 16x16 partial C tiles; diag adds too

    // Diagonal extraction. C layout: VGPR r holds M=r (lanes 0-15, N=lane)
    // and M=r+8 (lanes 16-31, N=lane-16). Edge m's score:
    //   m in 0..7  -> lane m,      component m
    //   m in 8..15 -> lane m+16,   component m-8
    const int s = lane & 7;
    float x0 = (s & 1) ? c[1] : c[0];
    float x1 = (s & 1) ? c[3] : c[2];
    float x2 = (s & 1) ? c[5] : c[4];
    float x3 = (s & 1) ? c[7] : c[6];
    float y0 = (s & 2) ? x1 : x0;
    float y1 = (s & 2) ? x3 : x2;
    float val = (s & 4) ? y1 : y0;

    if (lane < 8) {
        int m = lane;
        if (e0 + m < n_edges) out[e0 + m] = val;
    } else if (lane >= 24) {
        int m = lane - 16;
        if (e0 + m < n_edges) out[e0 + m] = val;
    }
}

extern "C" void kernel_launch(void* const* d_in, const int* in_sizes, int n_in,
                              void* d_out, int out_size, void* d_ws, size_t ws_size,
                              hipStream_t stream) {
    const float* h   = (const float*)d_in[0];
    const int*   src = (const int*)d_in[1];
    const int*   dst = (const int*)d_in[2];
    float*       out = (float*)d_out;

    const int n_edges = in_sizes[1];                       // 640000
    const int tiles   = (n_edges + EDGES_PER_WAVE - 1) / EDGES_PER_WAVE;
    const int blocks  = (tiles + WAVES_PER_BLOCK - 1) / WAVES_PER_BLOCK;

    edge_dot_wmma_f32<<<blocks, 256, 0, stream>>>(h, src, dst, out, n_edges);
}